// SmoothDecoder_14431090114810
// MI455X (gfx1250) — compile-verified
//
#include <hip/hip_runtime.h>
#include <hip/hip_bf16.h>
#include <math.h>

#define SIZE_U 6000
#define SIZE_V 6000
#define NN     (SIZE_U + SIZE_V)
#define KTOP   20
#define DD     256

typedef __attribute__((ext_vector_type(2))) float v2f;
typedef __attribute__((ext_vector_type(8))) float v8f;

// ---------------------------------------------------------------------------
// Kernel 1: per-row top-K selection + neighbor merge.
// One 256-thread block per row. The 48KB sim row is staged in LDS; top-20 is
// found by 20 block-wide argmax reductions (exact, index-tie-break like
// jax.lax.top_k); then thread d computes the weighted merge for feature dim d.
// Purely HBM-bandwidth bound (576MB sim read ~ 25us floor at 23.3 TB/s).
// ---------------------------------------------------------------------------
__global__ __launch_bounds__(256) void topk_merge_kernel(
    const float* __restrict__ u, const float* __restrict__ v,
    const float* __restrict__ sim, const unsigned char* __restrict__ mask,
    float* __restrict__ feat /* [NN x DD], u_s then v_s slots of d_out */)
{
    __shared__ float s_row[NN];
    __shared__ float s_rv[256];
    __shared__ int   s_ri[256];
    __shared__ float s_topv[KTOP];
    __shared__ int   s_topi[KTOP];

    const int n   = blockIdx.x;
    const int tid = threadIdx.x;
    const float* __restrict__ srow = sim + (size_t)n * NN;

    for (int i = tid; i < NN; i += 256) s_row[i] = srow[i];
    __syncthreads();

    for (int t = 0; t < KTOP; ++t) {
        float bv = -INFINITY; int bi = NN;
        for (int i = tid; i < NN; i += 256) {
            float vv = s_row[i];
            if (vv > bv || (vv == bv && i < bi)) { bv = vv; bi = i; }
        }
        s_rv[tid] = bv; s_ri[tid] = bi;
        __syncthreads();
        for (int off = 128; off > 0; off >>= 1) {
            if (tid < off) {
                float ov = s_rv[tid + off]; int oi = s_ri[tid + off];
                if (ov > s_rv[tid] || (ov == s_rv[tid] && oi < s_ri[tid])) {
                    s_rv[tid] = ov; s_ri[tid] = oi;
                }
            }
            __syncthreads();
        }
        if (tid == 0) {
            s_topv[t] = s_rv[0];
            s_topi[t] = s_ri[0];
            s_row[s_ri[0]] = -INFINITY;   // exclude for next round
        }
        __syncthreads();
    }

    // Weighted merge: thread `tid` owns feature dimension d = tid.
    const int d = tid;
    float acc = 0.f, wsum = 0.f;
    #pragma unroll
    for (int t = 0; t < KTOP; ++t) {
        float w  = s_topv[t];
        int   ix = s_topi[t];
        const float* __restrict__ f = (ix < SIZE_U)
            ? (u + (size_t)ix * DD)
            : (v + (size_t)(ix - SIZE_U) * DD);
        acc  += w * f[d];
        wsum += w;
    }
    const float* __restrict__ fo = (n < SIZE_U)
        ? (u + (size_t)n * DD)
        : (v + (size_t)(n - SIZE_U) * DD);
    float own = fo[d];
    float res = mask[n] ? own : (acc / wsum);
    feat[(size_t)n * DD + d] = res;
}

// ---------------------------------------------------------------------------
// Kernel 2: sigmoid(U @ V^T) with V_WMMA_F32_16X16X4_F32 (f32-exact path).
// 8 waves/block arranged 2(M) x 4(N); each wave owns a 64x64 output block as
// 4x4 WMMA subtiles (16 v8f accumulators). Operands (12MB total) are
// L2-resident (192MB L2) so fragments are loaded directly from global:
// per K-step of 4, 8 float2 loads feed 16 WMMAs (16 matrix ops / 64B / lane).
// A-fragment layout (32-bit 16x4, ISA 7.12.2): lanes 0-15 -> M=lane, K={0,1};
// lanes 16-31 -> M=lane-16, K={2,3}. B (4x16) mirrors it from row-major V.
// Edge tiles: clamp load rows (EXEC must stay all-1s for WMMA), guard stores.
// ---------------------------------------------------------------------------
__global__ __launch_bounds__(256) void gemm_sigmoid_kernel(
    const float* __restrict__ U, const float* __restrict__ V,
    float* __restrict__ out)
{
    const int lane  = threadIdx.x & 31;
    const int wid   = threadIdx.x >> 5;   // 0..7
    const int waveM = wid >> 2;           // 0..1
    const int waveN = wid & 3;            // 0..3
    const int half  = lane >> 4;          // 0: K=0,1  1: K=2,3
    const int lrow  = lane & 15;

    const int mBase = blockIdx.y * 128 + waveM * 64;
    const int nBase = blockIdx.x * 256 + waveN * 64;

    v8f acc[4][4];
    #pragma unroll
    for (int mt = 0; mt < 4; ++mt)
        #pragma unroll
        for (int nt = 0; nt < 4; ++nt)
            acc[mt][nt] = (v8f)0.f;

    const float* aPtr[4];
    const float* bPtr[4];
    #pragma unroll
    for (int mt = 0; mt < 4; ++mt) {
        int r = mBase + mt * 16 + lrow;
        if (r > SIZE_U - 1) r = SIZE_U - 1;          // clamp, keep EXEC full
        aPtr[mt] = U + (size_t)r * DD + 2 * half;
    }
    #pragma unroll
    for (int nt = 0; nt < 4; ++nt) {
        int r = nBase + nt * 16 + lrow;
        if (r > SIZE_V - 1) r = SIZE_V - 1;
        bPtr[nt] = V + (size_t)r * DD + 2 * half;
    }

    for (int k = 0; k < DD; k += 4) {
        v2f a[4], b[4];
        #pragma unroll
        for (int mt = 0; mt < 4; ++mt) a[mt] = *(const v2f*)(aPtr[mt] + k);
        #pragma unroll
        for (int nt = 0; nt < 4; ++nt) b[nt] = *(const v2f*)(bPtr[nt] + k);

        #pragma unroll
        for (int mt = 0; mt < 4; ++mt)
            #pragma unroll
            for (int nt = 0; nt < 4; ++nt)
                acc[mt][nt] = __builtin_amdgcn_wmma_f32_16x16x4_f32(
                    false, a[mt], false, b[nt],
                    (short)0, acc[mt][nt], false, false);
    }

    // C/D layout: VGPR r -> M = r (lanes 0-15) / r+8 (lanes 16-31), N = lrow.
    #pragma unroll
    for (int mt = 0; mt < 4; ++mt)
        #pragma unroll
        for (int nt = 0; nt < 4; ++nt)
            #pragma unroll
            for (int r = 0; r < 8; ++r) {
                int row = mBase + mt * 16 + r + half * 8;
                int col = nBase + nt * 16 + lrow;
                if (row < SIZE_U && col < SIZE_V) {
                    float x = acc[mt][nt][r];
                    out[(size_t)row * SIZE_V + col] = 1.f / (1.f + __expf(-x));
                }
            }
}

// ---------------------------------------------------------------------------
// d_in: [0]=u (6000x256 f32), [1]=v (6000x256 f32), [2]=sim (12000x12000 f32),
//       [3]=mask (12000 bool/u8)
// d_out: [sigmoid 6000x6000 | u_s 6000x256 | v_s 6000x256] f32
// ---------------------------------------------------------------------------
extern "C" void kernel_launch(void* const* d_in, const int* in_sizes, int n_in,
                              void* d_out, int out_size, void* d_ws, size_t ws_size,
                              hipStream_t stream) {
    (void)in_sizes; (void)n_in; (void)out_size; (void)d_ws; (void)ws_size;

    const float*         u    = (const float*)d_in[0];
    const float*         v    = (const float*)d_in[1];
    const float*         sim  = (const float*)d_in[2];
    const unsigned char* mask = (const unsigned char*)d_in[3];

    float* out  = (float*)d_out;
    float* feat = out + (size_t)SIZE_U * SIZE_V;          // u_s then v_s, contiguous

    topk_merge_kernel<<<NN, 256, 0, stream>>>(u, v, sim, mask, feat);

    dim3 grid((SIZE_V + 255) / 256, (SIZE_U + 127) / 128);
    gemm_sigmoid_kernel<<<grid, dim3(256), 0, stream>>>(
        feat, feat + (size_t)SIZE_U * DD, out);
}